// SetAbstraction_8323646619921
// MI455X (gfx1250) — compile-verified
//
#include <hip/hip_runtime.h>

#define NB   8
#define NPTS 4096
#define CIN  64
#define MC   1024
#define KN   64
#define NBM  (NB * MC)          // 8192 centroids
#define RR   (0.2f * 0.2f)

typedef float v2f __attribute__((ext_vector_type(2)));
typedef float v8f __attribute__((ext_vector_type(8)));
typedef unsigned long long u64;
typedef unsigned int u32;

__device__ __forceinline__ u64 umin64(u64 a, u64 b) { return a < b ? a : b; }
__device__ __forceinline__ u64 umax64(u64 a, u64 b) { return a > b ? a : b; }

// ---------------------------------------------------------------------------
// Farthest point sampling: one block per cloud. Sequential over M-1 picks,
// parallel distance update + argmax per step. Key packs (dist, ~idx) so that
// max-key == (max dist, lowest index) matching jnp.argmax tie-break.
// ---------------------------------------------------------------------------
__global__ __launch_bounds__(256) void fps_kernel(const float* __restrict__ pos,
                                                  float* __restrict__ centW) {
  __shared__ float px[NPTS], py[NPTS], pz[NPTS];
  __shared__ u64 wred[8];
  __shared__ int lastS;
  const int b = blockIdx.x, tid = threadIdx.x;
  const int lane = tid & 31, wid = tid >> 5;

  for (int i = tid; i < NPTS; i += 256) {
    const float* p = pos + (size_t)(b * NPTS + i) * 3;
    px[i] = p[0]; py[i] = p[1]; pz[i] = p[2];
  }
  float dl[16];
#pragma unroll
  for (int j = 0; j < 16; ++j) dl[j] = 1e10f;
  if (tid == 0) {
    lastS = 0;
    const float* p0 = pos + (size_t)(b * NPTS) * 3;
    float* c0 = centW + (size_t)(b * MC) * 3;
    c0[0] = p0[0]; c0[1] = p0[1]; c0[2] = p0[2];
  }
  __syncthreads();

  for (int m = 1; m < MC; ++m) {
    const int last = lastS;
    const float lx = px[last], ly = py[last], lz = pz[last];
    u64 best = 0ull;
#pragma unroll
    for (int j = 0; j < 16; ++j) {
      const int i = tid + j * 256;
      const float dx = px[i] - lx, dy = py[i] - ly, dz = pz[i] - lz;
      const float d = dx * dx + dy * dy + dz * dz;
      const float dm = fminf(dl[j], d);
      dl[j] = dm;
      const u64 key = ((u64)__float_as_uint(dm) << 32) | (u32)(0xFFFFFFFFu - (u32)i);
      best = umax64(best, key);
    }
    for (int off = 16; off; off >>= 1) best = umax64(best, __shfl_xor(best, off, 32));
    if (lane == 0) wred[wid] = best;
    __syncthreads();
    if (tid == 0) {
      u64 mx = wred[0];
      for (int w = 1; w < 8; ++w) mx = umax64(mx, wred[w]);
      const int widx = (int)(0xFFFFFFFFu - (u32)(mx & 0xFFFFFFFFull));
      lastS = widx;
      float* c = centW + (size_t)(b * MC + m) * 3;
      c[0] = px[widx]; c[1] = py[widx]; c[2] = pz[widx];
    }
    __syncthreads();
  }
}

// ---------------------------------------------------------------------------
// Ball query: one block per centroid. Filter candidates (d2 <= R^2) into LDS,
// then ONE-SHOT rank selection: keys (d2bits<<32)|idx are unique, so
// rank(my) = #{keys < my}; rank < 64 selects and orders exactly like
// top_k(-d2) (ascending d2, then ascending index). No iterative rounds.
// ---------------------------------------------------------------------------
__global__ __launch_bounds__(256) void ballq_kernel(const float* __restrict__ pos,
                                                    const float* __restrict__ centW,
                                                    int* __restrict__ nbrW,
                                                    int* __restrict__ cntW) {
  __shared__ u64 cand[1024];
  __shared__ int ccnt;
  const int bm = blockIdx.x, tid = threadIdx.x;
  const int b = bm >> 10;

  if (tid == 0) ccnt = 0;
  const float cx = centW[(size_t)bm * 3 + 0];
  const float cy = centW[(size_t)bm * 3 + 1];
  const float cz = centW[(size_t)bm * 3 + 2];
  __syncthreads();

  for (int j = 0; j < 16; ++j) {
    const int i = tid + j * 256;
    const float* p = pos + (size_t)(b * NPTS + i) * 3;
    const float dx = cx - p[0], dy = cy - p[1], dz = cz - p[2];
    const float d2 = dx * dx + dy * dy + dz * dz;
    if (d2 <= RR) {
      const int slot = atomicAdd(&ccnt, 1);
      if (slot < 1024) cand[slot] = ((u64)__float_as_uint(d2) << 32) | (u32)i;
    }
  }
  __syncthreads();
  const int nc = min(ccnt, 1024);
  const int nsel = min(nc, KN);

  for (int sidx = tid; sidx < nc; sidx += 256) {
    const u64 my = cand[sidx];
    int rank = 0;
    for (int j = 0; j < nc; ++j) rank += (cand[j] < my) ? 1 : 0;
    if (rank < KN) nbrW[(size_t)bm * KN + rank] = (int)(u32)(my & 0xFFFFFFFFull);
  }
  for (int r = nsel + tid; r < KN; r += 256) nbrW[(size_t)bm * KN + r] = 0;
  if (tid == 0) cntW[bm] = nsel;
}

// ---------------------------------------------------------------------------
// fp32 WMMA 16x16 tile GEMM: A (16 edges x K) from LDS, B = W[K][OUT] from
// global (L2-resident). Layouts per CDNA5 ISA 7.12.2 (32-bit A 16x4, C/D 16x16).
// ---------------------------------------------------------------------------
__device__ __forceinline__ v8f wmma_gemm_tile(const float* __restrict__ Abuf, int lds,
                                              const float* __restrict__ W,
                                              int Kloop, int Kreal, int OUT,
                                              int edgeBase, int colBase, int lane) {
  const int row = lane & 15;
  const int half = lane >> 4;
  const int kb = half * 2;
  const int col = colBase + row;
  const float* ar = Abuf + (size_t)(edgeBase + row) * lds;
  v8f acc = {0.f, 0.f, 0.f, 0.f, 0.f, 0.f, 0.f, 0.f};
  for (int k0 = 0; k0 < Kloop; k0 += 4) {
    const int ka = k0 + kb;
    v2f a;
    a.x = ar[ka];
    a.y = ar[ka + 1];
    v2f bb;
    bb.x = (ka < Kreal) ? W[(size_t)ka * OUT + col] : 0.f;
    bb.y = (ka + 1 < Kreal) ? W[(size_t)(ka + 1) * OUT + col] : 0.f;
    acc = __builtin_amdgcn_wmma_f32_16x16x4_f32(false, a, false, bb, (short)0, acc,
                                                false, false);
  }
  return acc;
}

// ---------------------------------------------------------------------------
// Fused per-centroid pass. MODE 1/2/3: run chain up to layer MODE, emit raw
// per-channel (sum,sumsq) partials for that layer's BN stats (deterministic
// fixed-order reduction). MODE 4: full chain with all BN params, masked max
// over K, writes outputs. h0 x-features staged via async global->LDS B128.
// ---------------------------------------------------------------------------
template <int MODE>
__global__ __launch_bounds__(256) void pass_kernel(
    const float* __restrict__ x, const float* __restrict__ pos,
    const float* __restrict__ W1, const float* __restrict__ W2,
    const float* __restrict__ W3, const float* __restrict__ centW,
    const int* __restrict__ nbrW, const int* __restrict__ cntW,
    const float* __restrict__ A1, const float* __restrict__ C1,
    const float* __restrict__ A2, const float* __restrict__ C2,
    const float* __restrict__ A3, const float* __restrict__ C3,
    float* __restrict__ pSum, float* __restrict__ pSq,
    float* __restrict__ xout, float* __restrict__ centOut,
    int* __restrict__ bselOut) {
  __shared__ float smem[64 * 68 + 64 * 132];  // A | B (h3 buffer overlaps B)
  __shared__ int nbrS[64];
  __shared__ float centS[3];
  __shared__ int cntS;
  __shared__ int maxI[128];
  float* Abuf = smem;
  float* Bbuf = smem + 64 * 68;
  float* Cbuf = smem + 64 * 68;  // 64 x 132 raw-h3 region (overlaps Bbuf, used after)

  const int bm = blockIdx.x, tid = threadIdx.x;
  const int b = bm >> 10;
  const int lane = tid & 31, wave = tid >> 5;
  const int half = lane >> 4, row16 = lane & 15;

  if (tid < 64) nbrS[tid] = nbrW[(size_t)bm * 64 + tid];
  if (tid < 3) centS[tid] = centW[(size_t)bm * 3 + tid];
  if (tid == 0) cntS = cntW[bm];
  if (MODE == 4 && tid < 128) maxI[tid] = 0;
  __syncthreads();
  const int cnt = cntS;

  // ---- stage h0 = [x_j, pos_j - cent] into Abuf (64 edges x 68) ----
  // x-part (cols 0..63): async global->LDS, 16-byte chunks, per-lane gather.
  {
    const float* xbase = x + (size_t)b * NPTS * CIN;
    for (int chunk = tid; chunk < 64 * 16; chunk += 256) {
      const int e = chunk >> 4;
      const int c4 = (chunk & 15) << 2;
      float* dstp = &Abuf[e * 68 + c4];
      const u32 ldsoff = (u32)(size_t)dstp;
      if (e < cnt) {
        const int p = nbrS[e];
        const int goff = (p * CIN + c4) * 4;  // byte offset from xbase
        asm volatile("global_load_async_to_lds_b128 %0, %1, %2 offset:0"
                     :
                     : "v"(ldsoff), "v"(goff), "s"(xbase)
                     : "memory");
      } else {
        float4 z = {0.f, 0.f, 0.f, 0.f};
        *(float4*)dstp = z;
      }
    }
    // pos-part + pad (cols 64..67)
    for (int i = tid; i < 64 * 4; i += 256) {
      const int e = i >> 2, c = 64 + (i & 3);
      float v = 0.f;
      if (e < cnt && c < 67) {
        const int p = nbrS[e];
        v = pos[(size_t)(b * NPTS + p) * 3 + (c - 64)] - centS[c - 64];
      }
      Abuf[e * 68 + c] = v;
    }
    asm volatile("s_wait_asynccnt 0x0" ::: "memory");
  }
  __syncthreads();

  // ---- layer 1: 67 -> 64 ----
  for (int t = wave; t < 16; t += 8) {
    const int es = t >> 2, ct = t & 3;
    v8f acc = wmma_gemm_tile(Abuf, 68, W1, 68, 67, 64, es * 16, ct * 16, lane);
    const int col = ct * 16 + row16;
    float a1 = 0.f, c1 = 0.f;
    if (MODE >= 2) { a1 = A1[col]; c1 = C1[col]; }
#pragma unroll
    for (int j = 0; j < 8; ++j) {
      const int e = es * 16 + half * 8 + j;
      float v = acc[j];
      if (MODE >= 2) v = fmaxf(fmaf(v, a1, c1), 0.f);
      Bbuf[e * 68 + col] = v;
    }
  }
  __syncthreads();
  if (MODE == 1) {
    if (tid < 64) {
      float s = 0.f, q = 0.f;
      for (int e = 0; e < cnt; ++e) { const float v = Bbuf[e * 68 + tid]; s += v; q += v * v; }
      pSum[(size_t)tid * NBM + bm] = s;
      pSq[(size_t)tid * NBM + bm] = q;
    }
    return;
  }

  // ---- layer 2: 64 -> 64 ----
  for (int t = wave; t < 16; t += 8) {
    const int es = t >> 2, ct = t & 3;
    v8f acc = wmma_gemm_tile(Bbuf, 68, W2, 64, 64, 64, es * 16, ct * 16, lane);
    const int col = ct * 16 + row16;
    float a2 = 0.f, c2 = 0.f;
    if (MODE >= 3) { a2 = A2[col]; c2 = C2[col]; }
#pragma unroll
    for (int j = 0; j < 8; ++j) {
      const int e = es * 16 + half * 8 + j;
      float v = acc[j];
      if (MODE >= 3) v = fmaxf(fmaf(v, a2, c2), 0.f);
      Abuf[e * 68 + col] = v;
    }
  }
  __syncthreads();
  if (MODE == 2) {
    if (tid < 64) {
      float s = 0.f, q = 0.f;
      for (int e = 0; e < cnt; ++e) { const float v = Abuf[e * 68 + tid]; s += v; q += v * v; }
      pSum[(size_t)tid * NBM + bm] = s;
      pSq[(size_t)tid * NBM + bm] = q;
    }
    return;
  }

  // ---- layer 3: 64 -> 128 ----
  for (int t = wave; t < 32; t += 8) {
    const int es = t >> 3, ct = t & 7;
    v8f acc = wmma_gemm_tile(Abuf, 68, W3, 64, 64, 128, es * 16, ct * 16, lane);
    const int col = ct * 16 + row16;
    if (MODE == 3) {
#pragma unroll
      for (int j = 0; j < 8; ++j) {
        const int e = es * 16 + half * 8 + j;
        Cbuf[e * 132 + col] = acc[j];
      }
    } else {
      const float a3 = A3[col], c3 = C3[col];
#pragma unroll
      for (int j = 0; j < 8; ++j) {
        const int e = es * 16 + half * 8 + j;
        if (e < cnt) {
          const float v = fmaxf(fmaf(acc[j], a3, c3), 0.f);
          atomicMax(&maxI[col], __float_as_int(v));  // v>=0: int bits monotone
        }
      }
    }
  }
  __syncthreads();
  if (MODE == 3) {
    if (tid < 128) {
      float s = 0.f, q = 0.f;
      for (int e = 0; e < cnt; ++e) { const float v = Cbuf[e * 132 + tid]; s += v; q += v * v; }
      pSum[(size_t)tid * NBM + bm] = s;
      pSq[(size_t)tid * NBM + bm] = q;
    }
    return;
  }

  // ---- MODE 4: outputs ----
  if (tid < 128) xout[(size_t)bm * 128 + tid] = __int_as_float(maxI[tid]);
  if (tid < 3) centOut[(size_t)bm * 3 + tid] = centS[tid];
  if (tid == 0) bselOut[bm] = b;
}

// ---------------------------------------------------------------------------
// Deterministic fixed-order reduction of per-block BN partials -> scale/shift.
// One block per output channel; double accumulation.
// ---------------------------------------------------------------------------
__global__ __launch_bounds__(256) void bn_reduce_kernel(const float* __restrict__ pSum,
                                                        const float* __restrict__ pSq,
                                                        const int* __restrict__ cntW,
                                                        const float* __restrict__ g,
                                                        const float* __restrict__ bb,
                                                        float* __restrict__ bnA,
                                                        float* __restrict__ bnC) {
  const int c = blockIdx.x, tid = threadIdx.x;
  __shared__ double rs[256], rq[256];
  __shared__ long long rc[256];
  double s = 0.0, q = 0.0;
  long long n = 0;
  const int base = tid * (NBM / 256);
  for (int i = base; i < base + (NBM / 256); ++i) {
    s += (double)pSum[(size_t)c * NBM + i];
    q += (double)pSq[(size_t)c * NBM + i];
    n += cntW[i];
  }
  rs[tid] = s; rq[tid] = q; rc[tid] = n;
  __syncthreads();
  for (int st = 128; st > 0; st >>= 1) {
    if (tid < st) { rs[tid] += rs[tid + st]; rq[tid] += rq[tid + st]; rc[tid] += rc[tid + st]; }
    __syncthreads();
  }
  if (tid == 0) {
    double cntTot = (double)rc[0];
    if (cntTot < 1.0) cntTot = 1.0;
    const double mu = rs[0] / cntTot;
    double var = rq[0] / cntTot - mu * mu;
    if (var < 0.0) var = 0.0;
    const double a = (double)g[c] / sqrt(var + 1e-5);
    bnA[c] = (float)a;
    bnC[c] = (float)((double)bb[c] - mu * a);
  }
}

// ---------------------------------------------------------------------------
extern "C" void kernel_launch(void* const* d_in, const int* in_sizes, int n_in,
                              void* d_out, int out_size, void* d_ws, size_t ws_size,
                              hipStream_t stream) {
  (void)in_sizes; (void)n_in; (void)out_size; (void)ws_size;
  const float* x   = (const float*)d_in[0];
  const float* pos = (const float*)d_in[1];
  const float* W1  = (const float*)d_in[3];
  const float* g1  = (const float*)d_in[4];
  const float* b1  = (const float*)d_in[5];
  const float* W2  = (const float*)d_in[6];
  const float* g2  = (const float*)d_in[7];
  const float* b2  = (const float*)d_in[8];
  const float* W3  = (const float*)d_in[9];
  const float* g3  = (const float*)d_in[10];
  const float* b3  = (const float*)d_in[11];

  float* wsf  = (float*)d_ws;
  float* centW = wsf;                               // NBM*3
  int*   nbrW  = (int*)(wsf + (size_t)NBM * 3);     // NBM*KN
  int*   cntW  = nbrW + (size_t)NBM * KN;           // NBM
  float* pSum  = (float*)(cntW + NBM);              // NBM*128
  float* pSq   = pSum + (size_t)NBM * 128;          // NBM*128
  float* A1    = pSq + (size_t)NBM * 128;
  float* C1    = A1 + 64;
  float* A2    = C1 + 64;
  float* C2    = A2 + 64;
  float* A3    = C2 + 64;
  float* C3    = A3 + 128;

  float* xout    = (float*)d_out;
  float* centOut = xout + (size_t)NBM * 128;
  int*   bselOut = (int*)(centOut + (size_t)NBM * 3);

  fps_kernel<<<NB, 256, 0, stream>>>(pos, centW);
  ballq_kernel<<<NBM, 256, 0, stream>>>(pos, centW, nbrW, cntW);

  pass_kernel<1><<<NBM, 256, 0, stream>>>(x, pos, W1, W2, W3, centW, nbrW, cntW,
                                          A1, C1, A2, C2, A3, C3, pSum, pSq,
                                          xout, centOut, bselOut);
  bn_reduce_kernel<<<64, 256, 0, stream>>>(pSum, pSq, cntW, g1, b1, A1, C1);

  pass_kernel<2><<<NBM, 256, 0, stream>>>(x, pos, W1, W2, W3, centW, nbrW, cntW,
                                          A1, C1, A2, C2, A3, C3, pSum, pSq,
                                          xout, centOut, bselOut);
  bn_reduce_kernel<<<64, 256, 0, stream>>>(pSum, pSq, cntW, g2, b2, A2, C2);

  pass_kernel<3><<<NBM, 256, 0, stream>>>(x, pos, W1, W2, W3, centW, nbrW, cntW,
                                          A1, C1, A2, C2, A3, C3, pSum, pSq,
                                          xout, centOut, bselOut);
  bn_reduce_kernel<<<128, 256, 0, stream>>>(pSum, pSq, cntW, g3, b3, A3, C3);

  pass_kernel<4><<<NBM, 256, 0, stream>>>(x, pos, W1, W2, W3, centW, nbrW, cntW,
                                          A1, C1, A2, C2, A3, C3, pSum, pSq,
                                          xout, centOut, bselOut);
}